// MyRNN_35450660061216
// MI455X (gfx1250) — compile-verified
//
#include <hip/hip_runtime.h>

typedef float v2f __attribute__((ext_vector_type(2)));
typedef float v8f __attribute__((ext_vector_type(8)));

#define HSZ 16
#define TLEN 1024

__device__ __forceinline__ float fast_rcp(float x) { return __builtin_amdgcn_rcpf(x); }
__device__ __forceinline__ float sigm(float x) { return fast_rcp(1.0f + __expf(-x)); }
__device__ __forceinline__ float ftanh(float x) {
    // tanh(x) = 1 - 2/(exp(2x)+1)
    return 1.0f - 2.0f * fast_rcp(1.0f + __expf(2.0f * x));
}

// One wave (32 lanes) per 16-batch tile. Hidden state kept transposed
// (h.T : [hidden, batch]) in WMMA C/D layout: vgpr j holds hidden=j in lanes
// 0-15 and hidden=j+8 in lanes 16-31, lane%16 = batch-in-tile.
__global__ __launch_bounds__(32) void gru_wmma_kernel(
    const float* __restrict__ x,     // [B, T, 1]
    const float* __restrict__ W_ih,  // [48, 1]
    const float* __restrict__ W_hh,  // [48, 16]
    const float* __restrict__ b_ih,  // [48]
    const float* __restrict__ b_hh,  // [48]
    const float* __restrict__ W_fc,  // [5, 16]
    const float* __restrict__ b_fc,  // [5]
    float* __restrict__ out)         // [B, 5]
{
    const int lane   = threadIdx.x;        // 0..31
    const int b0     = blockIdx.x * 16;    // batch tile base
    const int bl     = lane & 15;          // batch within tile (N index)
    const bool lohalf = (lane < 16);
    const int ghalf  = lohalf ? 0 : 8;     // C/D row offset for this lane half
    const int koff   = lohalf ? 0 : 2;     // A-layout K offset for this lane half

    // ---- Preload A = W_hh in 16x16x4 A-layout (constant across timesteps) ----
    // A tile for gate block blk, K-chunk kc: a[blk][kc][p] = W_hh[blk*16+M][4kc+p+koff]
    float a[3][4][2];
    #pragma unroll
    for (int blk = 0; blk < 3; ++blk)
        #pragma unroll
        for (int kc = 0; kc < 4; ++kc)
            #pragma unroll
            for (int p = 0; p < 2; ++p)
                a[blk][kc][p] = W_hh[(blk * 16 + bl) * HSZ + (4 * kc + p + koff)];

    // ---- Per-lane gate constants for C-operand init (gate row = j + ghalf) ----
    float wih[3][8], brz[2][8], bihn[8], bhhn[8];
    #pragma unroll
    for (int j = 0; j < 8; ++j) {
        const int gr = 0 * 16 + j + ghalf;
        const int gz = 1 * 16 + j + ghalf;
        const int gn = 2 * 16 + j + ghalf;
        wih[0][j] = W_ih[gr];
        wih[1][j] = W_ih[gz];
        wih[2][j] = W_ih[gn];
        brz[0][j] = b_ih[gr] + b_hh[gr];   // r-gate: biases fold together
        brz[1][j] = b_ih[gz] + b_hh[gz];   // z-gate: biases fold together
        bihn[j]   = b_ih[gn];              // n-gate: b_hh stays inside r*(...)
        bhhn[j]   = b_hh[gn];
    }

    const float* xp = x + (size_t)(b0 + bl) * TLEN;

    // h.T in C/D layout, start at zero
    float hc[8];
    #pragma unroll
    for (int j = 0; j < 8; ++j) hc[j] = 0.0f;

    float xv = xp[0];

    for (int t = 0; t < TLEN; ++t) {
        // prefetch next timestep's x
        const int tn = (t + 1 < TLEN) ? (t + 1) : (TLEN - 1);
        const float xnext = xp[tn];

        // ---- Build B = h.T in 4x16 B-layout chunks from C/D-layout hc ----
        // chunk kc, vgpr p: lanes0-15 -> K=4kc+p, lanes16-31 -> K=4kc+p+2
        float Bv[4][2];
        {
            const float s0 = __shfl_xor(hc[0], 16);
            const float s2 = __shfl_xor(hc[2], 16);
            Bv[0][0] = lohalf ? hc[0] : s2;
            Bv[2][0] = lohalf ? s0    : hc[2];
            const float s1 = __shfl_xor(hc[1], 16);
            const float s3 = __shfl_xor(hc[3], 16);
            Bv[0][1] = lohalf ? hc[1] : s3;
            Bv[2][1] = lohalf ? s1    : hc[3];
            const float s4 = __shfl_xor(hc[4], 16);
            const float s6 = __shfl_xor(hc[6], 16);
            Bv[1][0] = lohalf ? hc[4] : s6;
            Bv[3][0] = lohalf ? s4    : hc[6];
            const float s5 = __shfl_xor(hc[5], 16);
            const float s7 = __shfl_xor(hc[7], 16);
            Bv[1][1] = lohalf ? hc[5] : s7;
            Bv[3][1] = lohalf ? s5    : hc[7];
        }

        // ---- r gate: C init = x*W_ih + (b_ih+b_hh), then 4 chained WMMAs ----
        v8f c;
        #pragma unroll
        for (int j = 0; j < 8; ++j) c[j] = fmaf(xv, wih[0][j], brz[0][j]);
        #pragma unroll
        for (int kc = 0; kc < 4; ++kc) {
            v2f av; av.x = a[0][kc][0]; av.y = a[0][kc][1];
            v2f bv; bv.x = Bv[kc][0];   bv.y = Bv[kc][1];
            c = __builtin_amdgcn_wmma_f32_16x16x4_f32(false, av, false, bv,
                                                      (short)0, c, false, false);
        }
        float r[8];
        #pragma unroll
        for (int j = 0; j < 8; ++j) r[j] = sigm(c[j]);

        // ---- z gate ----
        #pragma unroll
        for (int j = 0; j < 8; ++j) c[j] = fmaf(xv, wih[1][j], brz[1][j]);
        #pragma unroll
        for (int kc = 0; kc < 4; ++kc) {
            v2f av; av.x = a[1][kc][0]; av.y = a[1][kc][1];
            v2f bv; bv.x = Bv[kc][0];   bv.y = Bv[kc][1];
            c = __builtin_amdgcn_wmma_f32_16x16x4_f32(false, av, false, bv,
                                                      (short)0, c, false, false);
        }
        float z[8];
        #pragma unroll
        for (int j = 0; j < 8; ++j) z[j] = sigm(c[j]);

        // ---- n gate: C init = b_hh only (gh_n), then n = tanh(gx_n + r*gh_n) ----
        #pragma unroll
        for (int j = 0; j < 8; ++j) c[j] = bhhn[j];
        #pragma unroll
        for (int kc = 0; kc < 4; ++kc) {
            v2f av; av.x = a[2][kc][0]; av.y = a[2][kc][1];
            v2f bv; bv.x = Bv[kc][0];   bv.y = Bv[kc][1];
            c = __builtin_amdgcn_wmma_f32_16x16x4_f32(false, av, false, bv,
                                                      (short)0, c, false, false);
        }
        #pragma unroll
        for (int j = 0; j < 8; ++j) {
            const float pre = fmaf(xv, wih[2][j], bihn[j]) + r[j] * c[j];
            const float n   = ftanh(pre);
            // h' = (1-z)*n + z*h = n + z*(h-n)
            hc[j] = fmaf(z[j], hc[j] - n, n);
        }

        xv = xnext;
    }

    // ---- Epilogue: out = tanh(h) @ W_fc.T + b_fc ----
    __shared__ float sh[16 * 16];  // [hidden][batch-in-tile]
    #pragma unroll
    for (int j = 0; j < 8; ++j)
        sh[(j + ghalf) * 16 + bl] = ftanh(hc[j]);
    __syncthreads();

    if (lane < 16) {
        #pragma unroll
        for (int o = 0; o < 5; ++o) {
            float acc = b_fc[o];
            #pragma unroll
            for (int k = 0; k < 16; ++k)
                acc = fmaf(sh[k * 16 + bl], W_fc[o * 16 + k], acc);
            out[(size_t)(b0 + bl) * 5 + o] = acc;
        }
    }
}

extern "C" void kernel_launch(void* const* d_in, const int* in_sizes, int n_in,
                              void* d_out, int out_size, void* d_ws, size_t ws_size,
                              hipStream_t stream) {
    const float* x    = (const float*)d_in[0];
    const float* W_ih = (const float*)d_in[1];
    const float* W_hh = (const float*)d_in[2];
    const float* b_ih = (const float*)d_in[3];
    const float* b_hh = (const float*)d_in[4];
    const float* W_fc = (const float*)d_in[5];
    const float* b_fc = (const float*)d_in[6];
    float* out = (float*)d_out;

    const int B = in_sizes[0] / TLEN;  // 4096
    dim3 grid(B / 16), block(32);
    gru_wmma_kernel<<<grid, block, 0, stream>>>(x, W_ih, W_hh, b_ih, b_hh,
                                                W_fc, b_fc, out);
}